// GPT_43937515438416
// MI455X (gfx1250) — compile-verified
//
#include <hip/hip_runtime.h>
#include <hip/hip_bf16.h>
#include <math.h>

typedef __attribute__((ext_vector_type(2))) float v2f;
typedef __attribute__((ext_vector_type(8))) float v8f;

// ---- problem dims (compile-time) ----
constexpr int Bc   = 2;
constexpr int Tc   = 2048;
constexpr int Dc   = 1024;
constexpr int Hc   = 16;
constexpr int HKVc = 4;
constexpr int DHc  = 64;
constexpr int HIDc = 4096;
constexpr int NTOK = Bc * Tc;          // 4096 token rows
constexpr float kEps = 1.1920929e-07f; // float32 eps

// ---------------- WMMA helper: fp32 16x16x4 ----------------
__device__ __forceinline__ v8f wmma4(v2f a, v2f b, v8f c) {
  // D = A(16x4 f32) x B(4x16 f32) + C(16x16 f32)
  return __builtin_amdgcn_wmma_f32_16x16x4_f32(false, a, false, b, (short)0, c,
                                               false, false);
}

// ---------------- gfx1250 async global->LDS copy (ASYNCcnt) ---------------
__device__ __forceinline__ void async_copy_b128(unsigned lds_addr,
                                                const void* gaddr) {
  asm volatile("global_load_async_to_lds_b128 %0, %1, off"
               :
               : "v"(lds_addr), "v"((unsigned long long)(uintptr_t)gaddr)
               : "memory");
}
__device__ __forceinline__ void wait_async_le4() {
  asm volatile("s_wait_asynccnt 0x4" ::: "memory");
}
__device__ __forceinline__ void wait_async_0() {
  asm volatile("s_wait_asynccnt 0x0" ::: "memory");
}

// ---------------- reductions (wave32) ----------------
__device__ __forceinline__ float wave_sum32(float v) {
#pragma unroll
  for (int o = 16; o > 0; o >>= 1) v += __shfl_xor(v, o);
  return v;
}
__device__ __forceinline__ float half_sum16(float v) {
#pragma unroll
  for (int o = 8; o > 0; o >>= 1) v += __shfl_xor(v, o);
  return v;
}
__device__ __forceinline__ float half_max16(float v) {
#pragma unroll
  for (int o = 8; o > 0; o >>= 1) v = fmaxf(v, __shfl_xor(v, o));
  return v;
}

// ---------------- fused residual-mix + RMSNorm ----------------
template <bool MIX>
__global__ void norm_kernel(const float* __restrict__ A,
                            const float* __restrict__ B,
                            const float* __restrict__ rm,
                            const float* __restrict__ lnw,
                            float* __restrict__ resid_out,
                            float* __restrict__ norm_out) {
  const int row = blockIdx.x;
  const int tid = threadIdx.x;
  const int d0  = tid * 4;               // 256 threads * 4 = 1024 = D
  const size_t base = (size_t)row * Dc + d0;

  float4 av = *(const float4*)(A + base);
  float r0, r1, r2, r3;
  if (MIX) {
    float4 bv = *(const float4*)(B + base);
    float4 m0 = *(const float4*)(rm + d0);
    float4 m1 = *(const float4*)(rm + Dc + d0);
    r0 = m0.x * av.x + m1.x * bv.x;
    r1 = m0.y * av.y + m1.y * bv.y;
    r2 = m0.z * av.z + m1.z * bv.z;
    r3 = m0.w * av.w + m1.w * bv.w;
  } else {
    r0 = av.x; r1 = av.y; r2 = av.z; r3 = av.w;
  }
  if (resid_out) {
    float4 rv = {r0, r1, r2, r3};
    *(float4*)(resid_out + base) = rv;
  }
  float ss = r0 * r0 + r1 * r1 + r2 * r2 + r3 * r3;

  __shared__ float red[8];
  float wsum = wave_sum32(ss);
  if ((tid & 31) == 0) red[tid >> 5] = wsum;
  __syncthreads();
  float tot = 0.f;
#pragma unroll
  for (int j = 0; j < 8; ++j) tot += red[j];

  const float s = rsqrtf(tot * (1.0f / (float)Dc) + kEps) * lnw[0];
  float4 ov = {r0 * s, r1 * s, r2 * s, r3 * s};
  *(float4*)(norm_out + base) = ov;
}

// --------- NT GEMM: async-LDS double-buffered, v_wmma_f32_16x16x4_f32 ------
// A:[M,K], W:[N,K] row-major (out = A @ W^T), C:[M,N].
// Block: 256 thr = 8 waves; block tile 64x64; wave tile 16x32 (A-frag reuse).
// K staged in KT=32 chunks via GLOBAL_LOAD_ASYNC_TO_LDS_B128 (double buffer).
// EPI 0: plain  EPI 1: (leaky_relu(v,0.5))^2  EPI 2: aux0[idx]+aux1[col]*v
template <int EPI>
__global__ void gemm_nt_kernel(const float* __restrict__ A,
                               const float* __restrict__ W,
                               float* __restrict__ C, int M, int N, int K,
                               const float* __restrict__ aux0,
                               const float* __restrict__ aux1) {
  constexpr int KT = 32;
  __shared__ float sA[2][64 * KT];
  __shared__ float sB[2][64 * KT];

  const int tid = threadIdx.x;
  const int w  = tid >> 5;
  const int l  = tid & 31;
  const int lj = l & 15;
  const int hi = l >> 4;
  const int wr = w & 3;    // 4 row tiles of 16  -> 64 rows
  const int wc = w >> 2;   // 2 col strips of 32 -> 64 cols
  const int bm = blockIdx.x * 64;
  const int bn = blockIdx.y * 64;

  // stage one 64xKT tile of A and of W into LDS buffer `buf` (async)
  auto issue = [&](int buf, int kt) {
#pragma unroll
    for (int rep = 0; rep < 2; ++rep) {
      const int c   = tid + rep * 256;   // 512 float4 chunks per 64x32 tile
      const int row = c >> 3;
      const int seg = (c & 7) * 4;
      async_copy_b128((unsigned)(uintptr_t)&sA[buf][row * KT + seg],
                      A + (size_t)(bm + row) * K + kt * KT + seg);
      async_copy_b128((unsigned)(uintptr_t)&sB[buf][row * KT + seg],
                      W + (size_t)(bn + row) * K + kt * KT + seg);
    }
  };

  const int nk = K / KT;
  issue(0, 0);

  v8f acc0 = {0.f, 0.f, 0.f, 0.f, 0.f, 0.f, 0.f, 0.f};
  v8f acc1 = {0.f, 0.f, 0.f, 0.f, 0.f, 0.f, 0.f, 0.f};

  for (int kt = 0; kt < nk; ++kt) {
    const int buf = kt & 1;
    if (kt + 1 < nk) {
      issue(buf ^ 1, kt + 1);  // prefetch next chunk (other buffer)
      wait_async_le4();        // previous 4 ops (this chunk) complete
    } else {
      wait_async_0();
    }
    __syncthreads();           // chunk visible to all waves

    const float* ab = &sA[buf][(wr * 16 + lj) * KT + 2 * hi];
    const float* b0 = &sB[buf][(wc * 32 + lj) * KT + 2 * hi];
    const float* b1 = &sB[buf][(wc * 32 + 16 + lj) * KT + 2 * hi];

    // batch all LDS fragment loads first, then a clean WMMA chain
    v2f af[8], bf0[8], bf1[8];
#pragma unroll
    for (int i = 0; i < 8; ++i) {
      af[i]  = *(const v2f*)(ab + i * 4);
      bf0[i] = *(const v2f*)(b0 + i * 4);
      bf1[i] = *(const v2f*)(b1 + i * 4);
    }
#pragma unroll
    for (int i = 0; i < 8; ++i) {
      acc0 = wmma4(af[i], bf0[i], acc0);
      acc1 = wmma4(af[i], bf1[i], acc1);
    }
    __syncthreads();           // done reading; next iter may refill this buf
  }

#pragma unroll
  for (int t = 0; t < 2; ++t) {
    const v8f& acc = t ? acc1 : acc0;
    const int col = bn + wc * 32 + t * 16 + lj;
#pragma unroll
    for (int r = 0; r < 8; ++r) {
      const int row = bm + wr * 16 + r + 8 * hi;
      const size_t idx = (size_t)row * N + col;
      float v = acc[r];
      if (EPI == 1) {
        float lr = v > 0.f ? v : 0.5f * v;
        C[idx] = lr * lr;
      } else if (EPI == 2) {
        C[idx] = aux0[idx] + aux1[col] * v;
      } else {
        C[idx] = v;
      }
    }
  }
}

// ---------------- per-head RMSNorm + RoPE + gain, v += ve ----------------
__global__ void qkv_post_kernel(float* __restrict__ Q, float* __restrict__ K,
                                float* __restrict__ V,
                                const float* __restrict__ VE,
                                const float* __restrict__ qgain) {
  const int n = blockIdx.x;        // token row
  const int t = n % Tc;            // position
  const int w = threadIdx.x >> 5;
  const int l = threadIdx.x & 31;
  const int j = l & 15;

  const float invf = __powf(10000.f, -(float)j / 16.f); // ROPE=32, half=16
  float sv, cv;
  __sincosf((float)t * invf, &sv, &cv);
  const float sgn = (l < 16) ? sv : -sv;

  for (int h = w; h < Hc; h += 8) {
    float* p = Q + (size_t)n * Dc + h * DHc;
    float f0 = p[l];       // dims 0..31 (rope region)
    float f1 = p[l + 32];  // dims 32..63 (pass-through)
    float ss = wave_sum32(f0 * f0 + f1 * f1);
    float r  = rsqrtf(ss * (1.f / (float)DHc) + kEps);
    f0 *= r; f1 *= r;
    float partner = __shfl_xor(f0, 16);
    f0 = f0 * cv + partner * sgn;
    const float g = qgain[h];
    p[l]      = f0 * g;
    p[l + 32] = f1 * g;
  }

  if (w < 4) {
    const int h = w;
    float* p = K + (size_t)n * (HKVc * DHc) + h * DHc;
    float f0 = p[l];
    float f1 = p[l + 32];
    float ss = wave_sum32(f0 * f0 + f1 * f1);
    float r  = rsqrtf(ss * (1.f / (float)DHc) + kEps);
    f0 *= r; f1 *= r;
    float partner = __shfl_xor(f0, 16);
    f0 = f0 * cv + partner * sgn;
    p[l]      = f0;
    p[l + 32] = f1;
  } else {
    const int h = w - 4;
    float* p = V + (size_t)n * (HKVc * DHc) + h * DHc;
    const float* e = VE + (size_t)n * (HKVc * DHc) + h * DHc;
    p[l]      += e[l];
    p[l + 32] += e[l + 32];
  }
}

// ---------------- flash attention + value-direction rejection -------------
__global__ void attn_kernel(const float* __restrict__ Q,
                            const float* __restrict__ Kr,
                            const float* __restrict__ Vr,
                            float* __restrict__ Y) {
  const int qt = blockIdx.x;       // query tile
  const int bh = blockIdx.y;
  const int b  = bh >> 4;
  const int hq = bh & 15;
  const int m  = hq >> 2;          // kv head (GRP=4)
  const int q0 = qt * 16;

  const int tid = threadIdx.x;
  const int w  = tid >> 5;
  const int l  = tid & 31;
  const int lj = l & 15;
  const int hi = l >> 4;

  __shared__ float sQ[16 * 64];
  __shared__ float sP[8][16 * 16];
  __shared__ float sMax[8][16];
  __shared__ float sSum[8][16];
  __shared__ float sAcc[8][16][64];
  __shared__ float sY[16 * 64];

  {
    const int e0 = tid * 4;
    const int row = e0 >> 6, col = e0 & 63;
    const float* src = Q + (size_t)(b * Tc + q0 + row) * Dc + hq * DHc + col;
    *(float4*)&sQ[e0] = *(const float4*)src;
  }
  __syncthreads();

  v2f qa[16];
#pragma unroll
  for (int i = 0; i < 16; ++i)
    qa[i] = *(const v2f*)&sQ[lj * 64 + i * 4 + 2 * hi];

  float rowmax[8], rowsum[8];
  v8f acc[4];
#pragma unroll
  for (int r = 0; r < 8; ++r) { rowmax[r] = -1e30f; rowsum[r] = 0.f; }
#pragma unroll
  for (int c = 0; c < 4; ++c) {
    v8f z = {0.f, 0.f, 0.f, 0.f, 0.f, 0.f, 0.f, 0.f};
    acc[c] = z;
  }

  for (int kt = w; kt <= qt; kt += 8) {
    // ---- batch-load K fragments (16x b64), then a clean 16-WMMA chain ----
    const float* kb =
        Kr + (size_t)(b * Tc + kt * 16 + lj) * (HKVc * DHc) + m * DHc + 2 * hi;
    v2f kfrag[16];
#pragma unroll
    for (int i = 0; i < 16; ++i) kfrag[i] = *(const v2f*)(kb + i * 4);

    v8f sc = {0.f, 0.f, 0.f, 0.f, 0.f, 0.f, 0.f, 0.f};
#pragma unroll
    for (int i = 0; i < 16; ++i) sc = wmma4(qa[i], kfrag[i], sc);

    // ---- prefetch V fragments; latency hidden under softmax VALU ----
    v2f vfrag[4][4];
#pragma unroll
    for (int i = 0; i < 4; ++i) {
      const int tok = b * Tc + kt * 16 + i * 4 + 2 * hi;
      const float* vb = Vr + (size_t)tok * (HKVc * DHc) + m * DHc;
#pragma unroll
      for (int c = 0; c < 4; ++c) {
        vfrag[i][c].x = vb[c * 16 + lj];
        vfrag[i][c].y = vb[(HKVc * DHc) + c * 16 + lj];
      }
    }

    // ---- online softmax update ----
    const int key = kt * 16 + lj;
#pragma unroll
    for (int r = 0; r < 8; ++r) {
      const int qrow = q0 + r + 8 * hi;
      float val = sc[r] * 0.125f;               // 1/sqrt(64)
      if (key > qrow) val = -1e30f;             // causal mask
      float mloc = half_max16(val);
      float mnew = fmaxf(rowmax[r], mloc);
      float sold = __expf(rowmax[r] - mnew);
      float p    = __expf(val - mnew);
      float psum = half_sum16(p);
      rowsum[r] = rowsum[r] * sold + psum;
      rowmax[r] = mnew;
#pragma unroll
      for (int c = 0; c < 4; ++c) acc[c][r] *= sold;
      sP[w][(r + 8 * hi) * 16 + lj] = p;        // re-layout C->A via LDS
    }

    // ---- acc += P . V (16 WMMAs, P frags from per-wave LDS) ----
#pragma unroll
    for (int i = 0; i < 4; ++i) {
      v2f pa = *(const v2f*)&sP[w][lj * 16 + i * 4 + 2 * hi];
#pragma unroll
      for (int c = 0; c < 4; ++c) acc[c] = wmma4(pa, vfrag[i][c], acc[c]);
    }
  }

  if (lj == 0) {
#pragma unroll
    for (int r = 0; r < 8; ++r) {
      sMax[w][r + 8 * hi] = rowmax[r];
      sSum[w][r + 8 * hi] = rowsum[r];
    }
  }
#pragma unroll
  for (int c = 0; c < 4; ++c)
#pragma unroll
    for (int r = 0; r < 8; ++r)
      sAcc[w][r + 8 * hi][c * 16 + lj] = acc[c][r];
  __syncthreads();

#pragma unroll
  for (int rep = 0; rep < 4; ++rep) {
    const int e = tid + rep * 256;
    const int row = e >> 6, col = e & 63;
    float M = -1e30f;
    for (int ww = 0; ww < 8; ++ww) M = fmaxf(M, sMax[ww][row]);
    float S = 0.f, num = 0.f;
    for (int ww = 0; ww < 8; ++ww) {
      const float f = __expf(sMax[ww][row] - M);
      S   += sSum[ww][row] * f;
      num += sAcc[ww][row][col] * f;
    }
    sY[e] = num / S;
  }
  __syncthreads();

#pragma unroll
  for (int rr = 0; rr < 2; ++rr) {
    const int row = w * 2 + rr;
    const float* vb =
        Vr + (size_t)(b * Tc + q0 + row) * (HKVc * DHc) + m * DHc;
    const float v0 = vb[l], v1 = vb[l + 32];
    const float y0 = sY[row * 64 + l], y1 = sY[row * 64 + l + 32];
    const float ssv = wave_sum32(v0 * v0 + v1 * v1);
    const float dp  = wave_sum32(y0 * v0 + y1 * v1);
    const float nrm = fmaxf(sqrtf(ssv), 1e-12f);
    const float coef = dp / (nrm * nrm);
    float* dst = Y + (size_t)(b * Tc + q0 + row) * Dc + hq * DHc;
    dst[l]      = y0 - coef * v0;
    dst[l + 32] = y1 - coef * v1;
  }
}

// ---------------- launch ----------------
extern "C" void kernel_launch(void* const* d_in, const int* in_sizes, int n_in,
                              void* d_out, int out_size, void* d_ws,
                              size_t ws_size, hipStream_t stream) {
  const float* x    = (const float*)d_in[0];
  const float* x0   = (const float*)d_in[1];
  const float* ve   = (const float*)d_in[2];
  const float* Wq   = (const float*)d_in[3];
  const float* Wk   = (const float*)d_in[4];
  const float* Wv   = (const float*)d_in[5];
  const float* Wo   = (const float*)d_in[6];
  const float* qg   = (const float*)d_in[7];
  const float* lnw  = (const float*)d_in[8];
  const float* ascl = (const float*)d_in[9];
  const float* mscl = (const float*)d_in[10];
  const float* rmix = (const float*)d_in[11];
  const float* Wfc  = (const float*)d_in[12];
  const float* Wmp  = (const float*)d_in[13];
  float* out = (float*)d_out;
  float* ws  = (float*)d_ws;

  const size_t ND  = (size_t)NTOK * Dc;
  const size_t NKV = (size_t)NTOK * HKVc * DHc;
  float* W0 = ws;        // mixed residual / post-attn residual
  float* W1 = W0 + ND;   // attn_in
  float* W2 = W1 + ND;   // q (then mlp_in)
  float* W3 = W2 + ND;   // k
  float* W4 = W3 + NKV;  // v
  float* W5 = W4 + NKV;  // attention output y
  float* W6 = W5 + ND;   // h*h [NTOK, HID]

  // 1) residual mix + rmsnorm
  norm_kernel<true><<<NTOK, 256, 0, stream>>>(x, x0, rmix, lnw, W0, W1);
  // 2) QKV projections (64x64 block tiles)
  gemm_nt_kernel<0><<<dim3(NTOK / 64, Dc / 64), 256, 0, stream>>>(
      W1, Wq, W2, NTOK, Dc, Dc, nullptr, nullptr);
  gemm_nt_kernel<0><<<dim3(NTOK / 64, (HKVc * DHc) / 64), 256, 0, stream>>>(
      W1, Wk, W3, NTOK, HKVc * DHc, Dc, nullptr, nullptr);
  gemm_nt_kernel<0><<<dim3(NTOK / 64, (HKVc * DHc) / 64), 256, 0, stream>>>(
      W1, Wv, W4, NTOK, HKVc * DHc, Dc, nullptr, nullptr);
  // 3) head rmsnorm + rope + gain, v += ve
  qkv_post_kernel<<<NTOK, 256, 0, stream>>>(W2, W3, W4, ve, qg);
  // 4) attention
  attn_kernel<<<dim3(Tc / 16, Bc * Hc), 256, 0, stream>>>(W2, W3, W4, W5);
  // 5) output proj, fused residual: W0 = W0 + attn_scale * (y @ Wo^T)
  gemm_nt_kernel<2><<<dim3(NTOK / 64, Dc / 64), 256, 0, stream>>>(
      W5, Wo, W0, NTOK, Dc, Dc, W0, ascl);
  // 6) mlp rmsnorm
  norm_kernel<false><<<NTOK, 256, 0, stream>>>(W0, nullptr, nullptr, lnw,
                                               nullptr, W2);
  // 7) fc + squared leaky relu
  gemm_nt_kernel<1><<<dim3(NTOK / 64, HIDc / 64), 256, 0, stream>>>(
      W2, Wfc, W6, NTOK, HIDc, Dc, nullptr, nullptr);
  // 8) down proj, fused residual into d_out
  gemm_nt_kernel<2><<<dim3(NTOK / 64, Dc / 64), 256, 0, stream>>>(
      W6, Wmp, out, NTOK, Dc, HIDc, W0, mscl);
}